// RNN_34626026340552
// MI455X (gfx1250) — compile-verified
//
#include <hip/hip_runtime.h>
#include <hip/hip_bf16.h>

typedef __attribute__((ext_vector_type(2))) float v2f;
typedef __attribute__((ext_vector_type(8))) float v8f;

namespace {

constexpr int T_STEPS = 5;
constexpr int BATCH   = 1000000;
constexpr int IN_DIM  = 5;
constexpr int HID     = 6;
constexpr int ELEMS_PER_BLOCK = 80;              // lcm(16 per wave, 5 per out-row)
constexpr int WAVES_PER_BLOCK = 5;
constexpr int THREADS = WAVES_PER_BLOCK * 32;    // wave32
constexpr int ROWS_PER_T = BATCH / T_STEPS;      // 200000 output rows per timestep

// One-instruction half-wave exchange: ds_swizzle_b32, group-of-32 mode,
// xor_mask=0x10, or_mask=0, and_mask=0x1f  ->  offset = (0x10<<10)|0x1f = 0x401F.
// All lanes are always active here, so the EXEC-off-reads-zero caveat never hits.
__device__ __forceinline__ float swap16(float v) {
  return __int_as_float(
      __builtin_amdgcn_ds_swizzle(__float_as_int(v), 0x401F));
}

__device__ __forceinline__ float tanh_fast(float x) {
#if __has_builtin(__builtin_amdgcn_tanhf)
  return __builtin_amdgcn_tanhf(x);              // gfx1250 v_tanh_f32 (1 TRANS op)
#else
  float e = __expf(-2.0f * __builtin_fabsf(x));  // v_exp_f32 path fallback
  float r = __fdividef(1.0f - e, 1.0f + e);
  return __builtin_copysignf(r, x);
#endif
}

__device__ __forceinline__ float sig_fast(float x) {
  return 0.5f * tanh_fast(0.5f * x) + 0.5f;
}

__device__ __forceinline__ v8f wmma4(v2f a, v2f b, v8f c) {
  // D = A(16x4,f32) * B(4x16,f32) + C(16x16,f32)  — exact f32 matrix op
  return __builtin_amdgcn_wmma_f32_16x16x4_f32(false, a, false, b, (short)0, c,
                                               false, false);
}

__device__ __forceinline__ void lstm_unit(float gi, float gf, float gg, float go,
                                          float& c, float& h) {
  float I = sig_fast(gi);
  float F = sig_fast(gf);
  float G = tanh_fast(gg);
  float O = sig_fast(go);
  c = F * c + I * G;
  h = O * tanh_fast(c);
}

// Concatenated-K weight columns. Layer0: z = [x(5) | h1(6) | pad0]; Layer1: z = [h1(6) | h2(6)].
__device__ __forceinline__ float wcol_l0(const float* wih, const float* whh, int R, int c) {
  if (c < 5)  return wih[R * 5 + c];
  if (c < 11) return whh[R * 6 + (c - 5)];
  return 0.0f;
}
__device__ __forceinline__ float wcol_l1(const float* wih, const float* whh, int R, int c) {
  if (c < 6) return wih[R * 6 + c];
  return whh[R * 6 + (c - 6)];
}

// Gate-row permutation. tile0: units 0..3 as (i,f,g,o) quads (rows 0-15).
// tile1: unit4 in rows 0-3, unit5 in rows 8-11, other rows zero -> each
// wave half ends up owning 3 complete gate quads for its element.
__device__ __forceinline__ int tile0_row(int m) {
  return (m & 3) * HID + (m >> 2);               // PyTorch order: i,f,g,o blocks
}
__device__ __forceinline__ bool tile1_row(int m, int& R) {
  if (m < 4)            { R = m * HID + 4;       return true; }
  if (m >= 8 && m < 12) { R = (m - 8) * HID + 5; return true; }
  return false;
}

} // namespace

__global__ __launch_bounds__(THREADS)
void lstm2_wmma_kernel(const float* __restrict__ x,
                       const float* __restrict__ w_ih0, const float* __restrict__ w_hh0,
                       const float* __restrict__ b_ih0, const float* __restrict__ b_hh0,
                       const float* __restrict__ w_ih1, const float* __restrict__ w_hh1,
                       const float* __restrict__ b_ih1, const float* __restrict__ b_hh1,
                       const float* __restrict__ w_out, const float* __restrict__ b_out,
                       float* __restrict__ out) {
  __shared__ float sp[T_STEPS * 2 * ELEMS_PER_BLOCK];  // [t][half][elem] partial dots

  const int tid  = threadIdx.x;
  const int wave = tid >> 5;
  const int lane = tid & 31;
  const int n    = lane & 15;            // WMMA column = element within wave tile
  const int hi   = lane >> 4;            // wave half (selects K-pair / row block)
  const int e    = wave * 16 + n;        // element index within block (0..79)
  const int b    = blockIdx.x * ELEMS_PER_BLOCK + e;

  // ---- A fragments (wave-uniform weights). k-step s covers z-cols 4s..4s+3;
  //      A layout: lane holds row (lane&15), cols 4s+2*hi and 4s+2*hi+1. ----
  v2f A0t0[3], A0t1[3], A1t0[3], A1t1[3];
  {
    const int m  = n;
    const int R0 = tile0_row(m);
    int R1; const bool v1 = tile1_row(m, R1);
#pragma unroll
    for (int s = 0; s < 3; ++s) {
      const int c = 4 * s + 2 * hi;
      A0t0[s][0] = wcol_l0(w_ih0, w_hh0, R0, c);
      A0t0[s][1] = wcol_l0(w_ih0, w_hh0, R0, c + 1);
      A0t1[s][0] = v1 ? wcol_l0(w_ih0, w_hh0, R1, c)     : 0.0f;
      A0t1[s][1] = v1 ? wcol_l0(w_ih0, w_hh0, R1, c + 1) : 0.0f;
      A1t0[s][0] = wcol_l1(w_ih1, w_hh1, R0, c);
      A1t0[s][1] = wcol_l1(w_ih1, w_hh1, R0, c + 1);
      A1t1[s][0] = v1 ? wcol_l1(w_ih1, w_hh1, R1, c)     : 0.0f;
      A1t1[s][1] = v1 ? wcol_l1(w_ih1, w_hh1, R1, c + 1) : 0.0f;
    }
  }

  // ---- fold biases into the WMMA C initializer (C layout: VGPR r <-> row r + 8*hi) ----
  v8f c0i0 = {}, c1i0 = {}, c0i1 = {}, c1i1 = {};
#pragma unroll
  for (int r = 0; r < 8; ++r) {
    const int m = r + 8 * hi;
    const int R = tile0_row(m);
    c0i0[r] = b_ih0[R] + b_hh0[R];
    c0i1[r] = b_ih1[R] + b_hh1[R];
    int R1;
    if (tile1_row(m, R1)) {
      c1i0[r] = b_ih0[R1] + b_hh0[R1];
      c1i1[r] = b_ih1[R1] + b_hh1[R1];
    }
  }

  // ---- per-lane output-projection coefficients (out row mixes 5 consecutive b) ----
  const int p  = e % 5;                          // position within the 5-element group
  const int u0 = hi ? 2 : 0, u1 = hi ? 3 : 1, u2 = hi ? 5 : 4;  // this lane's units
  const float wl0 = w_out[p * HID + u0];
  const float wl1 = w_out[p * HID + u1];
  const float wl2 = w_out[p * HID + u2];

  // ---- preload all x features this lane needs (low half: feats 0,1,4; high: 2,3) ----
  float xA[T_STEPS], xB[T_STEPS], xC[T_STEPS];
#pragma unroll
  for (int t = 0; t < T_STEPS; ++t) {
    const float* xp = x + ((size_t)t * BATCH + b) * IN_DIM;
    xA[t] = xp[2 * hi];
    xB[t] = xp[2 * hi + 1];
    xC[t] = xp[4];
  }

  // ---- recurrence; lane owns 3 hidden units per layer (low: 0,1,4 / high: 2,3,5) ----
  float h1[3] = {0.f, 0.f, 0.f}, c1s[3] = {0.f, 0.f, 0.f};
  float h2[3] = {0.f, 0.f, 0.f}, c2s[3] = {0.f, 0.f, 0.f};

#pragma unroll
  for (int t = 0; t < T_STEPS; ++t) {
    // ================= layer 0: gates = [Wih0|Whh0] * [x_t|h1] + bias =================
    const float sw1 = swap16(h1[1]);             // exchange slot-1 h1 across halves
    v2f B0, B1, B2;                              // B frag: V0 = rows {0,2}, V1 = rows {1,3}
    B0[0] = xA[t];               B0[1] = xB[t];                  // rows 0-3: x0..x3
    B1[0] = hi ? sw1 : xC[t];    B1[1] = h1[0];                  // rows 4-7: x4,h1_0..h1_2
    B2[0] = hi ? h1[2] : sw1;    B2[1] = hi ? 0.0f : h1[2];      // rows 8-11: h1_3..h1_5,0

    v8f g0 = c0i0, g1 = c1i0;
    g0 = wmma4(A0t0[0], B0, g0);  g0 = wmma4(A0t0[1], B1, g0);  g0 = wmma4(A0t0[2], B2, g0);
    g1 = wmma4(A0t1[0], B0, g1);  g1 = wmma4(A0t1[1], B1, g1);  g1 = wmma4(A0t1[2], B2, g1);

    lstm_unit(g0[0], g0[1], g0[2], g0[3], c1s[0], h1[0]);
    lstm_unit(g0[4], g0[5], g0[6], g0[7], c1s[1], h1[1]);
    lstm_unit(g1[0], g1[1], g1[2], g1[3], c1s[2], h1[2]);

    // ================= layer 1: gates = [Wih1|Whh1] * [h1|h2] + bias =================
    const float ySw2 = swap16(h1[2]);            // y5 for the low half
    const float s0 = swap16(h2[0]);
    const float s1 = swap16(h2[1]);
    const float s2 = swap16(h2[2]);
    v2f D0, D1, D2;
    D0[0] = h1[0];            D0[1] = h1[1];                 // rows 0-3: y0..y3
    D1[0] = hi ? s0 : h1[2];  D1[1] = hi ? s1 : ySw2;        // rows 4-7: y4,y5,h2_0,h2_1
    D2[0] = hi ? s2 : s0;     D2[1] = hi ? h2[2] : s1;       // rows 8-11: h2_2..h2_5

    v8f q0 = c0i1, q1 = c1i1;
    q0 = wmma4(A1t0[0], D0, q0);  q0 = wmma4(A1t0[1], D1, q0);  q0 = wmma4(A1t0[2], D2, q0);
    q1 = wmma4(A1t1[0], D0, q1);  q1 = wmma4(A1t1[1], D1, q1);  q1 = wmma4(A1t1[2], D2, q1);

    lstm_unit(q0[0], q0[1], q0[2], q0[3], c2s[0], h2[0]);
    lstm_unit(q0[4], q0[5], q0[6], q0[7], c2s[1], h2[1]);
    lstm_unit(q1[0], q1[1], q1[2], q1[3], c2s[2], h2[2]);

    // partial of the output projection for this element at this timestep
    sp[(t * 2 + hi) * ELEMS_PER_BLOCK + e] = h2[0] * wl0 + h2[1] * wl1 + h2[2] * wl2;
  }

  __syncthreads();

  // ---- combine: out[t*200000 + blk*16 + r] = b_out + sum over 5 elems x 2 halves ----
  if (tid < T_STEPS * 16) {
    const int t   = tid >> 4;
    const int r16 = tid & 15;
    float sum = b_out[0];
#pragma unroll
    for (int pp = 0; pp < 5; ++pp) {
      const int ee = r16 * 5 + pp;
      sum += sp[(t * 2 + 0) * ELEMS_PER_BLOCK + ee] +
             sp[(t * 2 + 1) * ELEMS_PER_BLOCK + ee];
    }
    out[t * ROWS_PER_T + blockIdx.x * 16 + r16] = sum;
  }
}

extern "C" void kernel_launch(void* const* d_in, const int* in_sizes, int n_in,
                              void* d_out, int out_size, void* d_ws, size_t ws_size,
                              hipStream_t stream) {
  const float* x     = (const float*)d_in[0];
  const float* w_ih0 = (const float*)d_in[1];
  const float* w_hh0 = (const float*)d_in[2];
  const float* b_ih0 = (const float*)d_in[3];
  const float* b_hh0 = (const float*)d_in[4];
  const float* w_ih1 = (const float*)d_in[5];
  const float* w_hh1 = (const float*)d_in[6];
  const float* b_ih1 = (const float*)d_in[7];
  const float* b_hh1 = (const float*)d_in[8];
  const float* w_out = (const float*)d_in[9];
  const float* b_out = (const float*)d_in[10];
  float* out = (float*)d_out;

  const int grid = BATCH / ELEMS_PER_BLOCK;  // 12500 blocks of 5 waves
  lstm2_wmma_kernel<<<grid, THREADS, 0, stream>>>(
      x, w_ih0, w_hh0, b_ih0, b_hh0, w_ih1, w_hh1, b_ih1, b_hh1, w_out, b_out, out);
}